// XLayer_b_90340342104851
// MI455X (gfx1250) — compile-verified
//
#include <hip/hip_runtime.h>

// Per-sample 3x3 SAME conv, NHWC fp32, implicit GEMM on V_WMMA_F32_16X16X4_F32
// (CDNA5 / gfx1250, wave32).
//
//  - 2x2 register tiling: each wave computes 32 pixels x 32 channels (4 acc),
//    each A/B fragment pair feeds 4 WMMAs.
//  - Bank-conflict-free LDS:
//      sX pixel stride padded 32 -> 36 floats: lane m banks (36m+k0)%64 are
//      16 distinct values; hi half (k0+2) uses complementary mod-4 classes.
//      sW3[tap][cin/4][f][4] quad-interleaved: fragment (k0,k0+1) is one
//      contiguous 8B word at banks (4n+2hi)%64 -> lo/hi halves disjoint.
//
// Shapes: x (32,128,128,32), params (32, 9*32*32 + 32), out (32,128,128,32).

typedef __attribute__((ext_vector_type(2))) float v2f;
typedef __attribute__((ext_vector_type(8))) float v8f;

#define CIN   32
#define FOUT  32
#define Hdim  128
#define Wdim  128
#define KHW   3
#define TILE_W 32
#define ROWS_PER_BLK 4
#define XROWS (ROWS_PER_BLK + 2)         // 6 input rows incl. halo
#define XCOLS (TILE_W + 2)               // 34 input cols incl. halo
#define XPIX_STRIDE 36                   // padded cin stride (floats), 9 float4
#define WSIZE (KHW * KHW * CIN * FOUT)   // 9216
#define PSTRIDE (WSIZE + FOUT)           // 9248 floats per batch row

__global__ __launch_bounds__(128)
void conv3x3_wmma_f32(const float* __restrict__ x,
                      const float* __restrict__ params,
                      float* __restrict__ out) {
    // sW3 flat index: tap*1024 + (cin/4)*128 + f*4 + (cin%4)
    __shared__ float sW3[WSIZE];                            // 36 KB
    __shared__ float sX[XROWS * XCOLS * XPIX_STRIDE];       // 28.7 KB (padded)
    __shared__ float sB[FOUT];

    const int tid  = threadIdx.x;
    const int b    = blockIdx.y;
    const int wt   = blockIdx.x & 3;            // 32-wide column tile
    const int hblk = blockIdx.x >> 2;           // 4-row block
    const int h0   = hblk * ROWS_PER_BLK;
    const int w0   = wt * TILE_W;

    // ---- weights: gather-repack global->LDS, quad-interleaved layout ----
    // slot j (float4 of sW3): tap=j>>8, q=(j>>5)&7, f=j&31
    // dst float4[rr] = W[4q+rr][f] = pb[(tap*32 + 4q + rr)*32 + f]
    {
        const float* pb = params + (size_t)b * PSTRIDE;
        float4* wdst = (float4*)sW3;
        #pragma unroll
        for (int it = 0; it < WSIZE / 4 / 128; ++it) {      // 18 iterations
            const int j   = tid + 128 * it;
            const int tap = j >> 8;
            const int q   = (j >> 5) & 7;
            const int f   = j & 31;
            const float* src = pb + ((tap * 32 + 4 * q) * 32 + f);
            wdst[j] = make_float4(src[0], src[32], src[64], src[96]);
        }
        if (tid < FOUT)
            sB[tid] = pb[WSIZE + tid];
    }

    // ---- input tile with zero padding: 6 rows x 34 cols x 32 ch (stride 36) ----
    const int NXPIX = XROWS * XCOLS;            // 204 pixels
    float4* xdst = (float4*)sX;
    for (int i = tid; i < NXPIX * 8; i += 128) {
        const int pix = i >> 3;                 // 0..203
        const int c4  = i & 7;                  // which float4 of the 8 used
        const int r   = pix / XCOLS;
        const int c   = pix - r * XCOLS;
        const int row = h0 - 1 + r;
        const int col = w0 - 1 + c;
        float4 v = make_float4(0.f, 0.f, 0.f, 0.f);
        if ((unsigned)row < (unsigned)Hdim && (unsigned)col < (unsigned)Wdim)
            v = *(const float4*)(x + ((((size_t)b * Hdim + row) * Wdim + col) * CIN) + c4 * 4);
        xdst[pix * (XPIX_STRIDE / 4) + c4] = v; // stride 9 float4, last unused
    }
    __syncthreads();

    // ---- per-wave: output row h0+wave, 32 pixels x 32 channels ----
    const int lane  = tid & 31;
    const int wave  = tid >> 5;                 // 0..3 -> output row
    const int mlane = lane & 15;                // A: M row / B,D: N column
    const int hi    = lane >> 4;                // K pair select (A/B), M half (D)

    v8f acc00 = {}, acc01 = {}, acc10 = {}, acc11 = {};

    #pragma unroll
    for (int kh = 0; kh < KHW; ++kh) {
        #pragma unroll
        for (int kw = 0; kw < KHW; ++kw) {
            // A: input row (wave+kh), pixel (mlane + kw) [+16 for tile 1]
            const float* xrow = &sX[((kh + wave) * XCOLS + mlane + kw) * XPIX_STRIDE];
            // B: tap*1024 + (kb/4)*128 + f*4 + 2*hi ; f = mlane [+16 tile 1]
            const float* wb   = &sW3[(kh * 3 + kw) * 1024 + mlane * 4 + 2 * hi];
            #pragma unroll
            for (int kb = 0; kb < CIN; kb += 4) {
                const int k0 = kb + 2 * hi;
                v2f a0 = { xrow[k0],                       xrow[k0 + 1] };
                v2f a1 = { xrow[16 * XPIX_STRIDE + k0],    xrow[16 * XPIX_STRIDE + k0 + 1] };
                const float* bp = wb + kb * 32;            // (kb/4)*128
                v2f b0 = *(const v2f*)bp;                  // f = mlane
                v2f b1 = *(const v2f*)(bp + 64);           // f = mlane + 16
                acc00 = __builtin_amdgcn_wmma_f32_16x16x4_f32(false, a0, false, b0,
                        (short)0, acc00, false, false);
                acc01 = __builtin_amdgcn_wmma_f32_16x16x4_f32(false, a0, false, b1,
                        (short)0, acc01, false, false);
                acc10 = __builtin_amdgcn_wmma_f32_16x16x4_f32(false, a1, false, b0,
                        (short)0, acc10, false, false);
                acc11 = __builtin_amdgcn_wmma_f32_16x16x4_f32(false, a1, false, b1,
                        (short)0, acc11, false, false);
            }
        }
    }

    // ---- bias + store. D layout: VGPR r -> pixel (r + 8*hi), chan mlane ----
    const float bias0 = sB[mlane];
    const float bias1 = sB[16 + mlane];
    const int   h     = h0 + wave;
    float* obase = out
        + ((((size_t)b * Hdim + h) * Wdim) + w0 + 8 * hi) * FOUT + mlane;
    #pragma unroll
    for (int r = 0; r < 8; ++r) {
        obase[(size_t)r * FOUT]             = acc00[r] + bias0;   // px r+8hi,    f mlane
        obase[(size_t)r * FOUT + 16]        = acc01[r] + bias1;   // px r+8hi,    f mlane+16
        obase[(size_t)(16 + r) * FOUT]      = acc10[r] + bias0;   // px 16+r+8hi, f mlane
        obase[(size_t)(16 + r) * FOUT + 16] = acc11[r] + bias1;   // px 16+r+8hi, f mlane+16
    }
}

extern "C" void kernel_launch(void* const* d_in, const int* in_sizes, int n_in,
                              void* d_out, int out_size, void* d_ws, size_t ws_size,
                              hipStream_t stream) {
    (void)in_sizes; (void)n_in; (void)d_ws; (void)ws_size; (void)out_size;
    const float* x      = (const float*)d_in[0];
    const float* params = (const float*)d_in[1];
    float* out          = (float*)d_out;

    // grid.x = (H / ROWS_PER_BLK) * (W / TILE_W) = 32 * 4 = 128 ; grid.y = 32
    dim3 grid((Hdim / ROWS_PER_BLK) * (Wdim / TILE_W), 32);
    conv3x3_wmma_f32<<<grid, 128, 0, stream>>>(x, params, out);
}